// LocalMHARefiner_1443109011707
// MI455X (gfx1250) — compile-verified
//
#include <hip/hip_runtime.h>
#include <hip/hip_bf16.h>

typedef __attribute__((ext_vector_type(2))) float v2f;
typedef __attribute__((ext_vector_type(8))) float v8f;

#define HH 128
#define WW 256
#define NN (HH * WW)          // 32768 pixels per batch
#define BB 4
#define CF 80
#define EMB 64

// ---------------------------------------------------------------------------
// Stage 1: Q/K/V projection via fp32 WMMA.
//   Out[b, e, n] = sum_c W[e, c] * feat[b, c, n]
// One wave computes one 16(e) x 16(n) tile of one of {Q,K,V} for one batch,
// accumulating over K=80 in 20 steps of V_WMMA_F32_16X16X4_F32.
// A-matrix (16x4 f32) layout: lanes 0-15 row M=lane, v0=K0,v1=K1;
//                             lanes16-31 row M=lane-16, v0=K2,v1=K3.
// B-matrix (4x16) mirrored; C/D 16x16: VGPR g -> M=g (lanes0-15) / M=g+8.
// ---------------------------------------------------------------------------
__global__ void qkv_wmma(const float* __restrict__ feat,
                         const float* __restrict__ Wq,
                         const float* __restrict__ Wk,
                         const float* __restrict__ Wv,
                         float* __restrict__ Q,
                         float* __restrict__ K,
                         float* __restrict__ V) {
  const int wave = (blockIdx.x << 3) | (threadIdx.x >> 5);  // 8 waves / 256-thr block
  const int lane = threadIdx.x & 31;

  const int ntile = wave & 2047;      // NN/16 = 2048
  int rest = wave >> 11;
  const int etile = rest & 3;         // EMB/16 = 4
  rest >>= 2;
  const int mat = rest % 3;           // 0=Q 1=K 2=V
  const int b   = rest / 3;

  const float* Wm  = (mat == 0) ? Wq : (mat == 1) ? Wk : Wv;
  float*       Out = (mat == 0) ? Q  : (mat == 1) ? K  : V;

  const int n0  = ntile << 4;
  const int m0  = etile << 4;
  const int low = lane & 15;          // row (A) / col (B,D)
  const int hi  = lane >> 4;          // 0 or 1

  const float* featB = feat + (size_t)b * CF * NN + n0 + low;
  const float* Wrow  = Wm + (size_t)(m0 + low) * CF;

  v8f acc = {};
  for (int k0 = 0; k0 < CF; k0 += 4) {
    const int ka = k0 + (hi << 1);
    v2f a, bmat;
    a[0] = Wrow[ka];
    a[1] = Wrow[ka + 1];
    bmat[0] = featB[(size_t)ka * NN];
    bmat[1] = featB[(size_t)(ka + 1) * NN];
    acc = __builtin_amdgcn_wmma_f32_16x16x4_f32(
        false, a, false, bmat, (short)0, acc, false, false);
  }

  float* OutB = Out + ((size_t)b * EMB + m0) * NN + n0 + low;
  const int rowHalf = hi << 3;
#pragma unroll
  for (int g = 0; g < 8; ++g)
    OutB[(size_t)(g + rowHalf) * NN] = acc[g];
}

// ---------------------------------------------------------------------------
// Stage 2: local 3x3 window attention with the reference's channel-mixing
// unfold layout:  f = j*64 + (h*16+d)  ->  channel c = f/9, winpos wp = f%9.
// One thread per pixel. All (c,wp) fold to constants after unrolling, so each
// K/V access is one global_load with an immediate offset off 9 clamped window
// pointers, masked by 0/1 for out-of-range (zero padding).
// ---------------------------------------------------------------------------
__global__ void attn_win(const float* __restrict__ Qb,
                         const float* __restrict__ Kb,
                         const float* __restrict__ Vb,
                         const float* __restrict__ Wo,
                         float* __restrict__ delta) {
  const int idx = blockIdx.x * blockDim.x + threadIdx.x;  // b*NN + n
  const int b = idx >> 15;       // NN = 2^15
  const int n = idx & (NN - 1);
  const int y = n >> 8;          // WW = 256
  const int x = n & 255;

  const float* Kp[9];
  const float* Vp[9];
  float msk[9];
#pragma unroll
  for (int wp = 0; wp < 9; ++wp) {
    const int dy = wp / 3 - 1, dx = wp % 3 - 1;
    const int yy = y + dy, xx = x + dx;
    const bool ok = ((unsigned)yy < HH) && ((unsigned)xx < WW);
    const int yc = min(max(yy, 0), HH - 1);
    const int xc = min(max(xx, 0), WW - 1);
    const size_t off = (size_t)b * EMB * NN + (size_t)yc * WW + xc;
    Kp[wp] = Kb + off;
    Vp[wp] = Vb + off;
    msk[wp] = ok ? 1.0f : 0.0f;
  }

  float dacc[8];
#pragma unroll
  for (int a = 0; a < 8; ++a) dacc[a] = 0.0f;

  const float* Qn = Qb + (size_t)b * EMB * NN + n;
  const float SCALE = 0.25f;  // HEAD_DIM^-0.5, HEAD_DIM=16

#pragma unroll
  for (int h = 0; h < 4; ++h) {
    float q[16];
#pragma unroll
    for (int d = 0; d < 16; ++d) q[d] = Qn[(size_t)(h * 16 + d) * NN];

    float sc[9];
#pragma unroll
    for (int j = 0; j < 9; ++j) {
      float s = 0.0f;
#pragma unroll
      for (int d = 0; d < 16; ++d) {
        const int f = j * 64 + h * 16 + d;
        const int c = f / 9, wp = f % 9;
        s += q[d] * (Kp[wp][(size_t)c * NN] * msk[wp]);
      }
      sc[j] = s * SCALE;
    }

    // softmax over the 9 window positions
    float mx = sc[0];
#pragma unroll
    for (int j = 1; j < 9; ++j) mx = fmaxf(mx, sc[j]);
    float sum = 0.0f;
#pragma unroll
    for (int j = 0; j < 9; ++j) { sc[j] = __expf(sc[j] - mx); sum += sc[j]; }
    const float inv = 1.0f / sum;
#pragma unroll
    for (int j = 0; j < 9; ++j) sc[j] *= inv;

    // context
    float ctx[16];
#pragma unroll
    for (int d = 0; d < 16; ++d) ctx[d] = 0.0f;
#pragma unroll
    for (int j = 0; j < 9; ++j) {
#pragma unroll
      for (int d = 0; d < 16; ++d) {
        const int f = j * 64 + h * 16 + d;
        const int c = f / 9, wp = f % 9;
        ctx[d] += sc[j] * (Vp[wp][(size_t)c * NN] * msk[wp]);
      }
    }

    // partial output projection: dacc[a] += Wo[a, h*16+d] * ctx[d]
#pragma unroll
    for (int a = 0; a < 8; ++a) {
      float s = 0.0f;
#pragma unroll
      for (int d = 0; d < 16; ++d) s += Wo[a * EMB + h * 16 + d] * ctx[d];
      dacc[a] += s;
    }
  }

#pragma unroll
  for (int a = 0; a < 8; ++a)
    delta[((size_t)b * 8 + a) * NN + n] = dacc[a];
}

// ---------------------------------------------------------------------------
// Stage 3: out = attn_vol + (disp_gap >= 2 ? 0 : delta), broadcast over D=48.
// float4-vectorized pure streaming kernel (the roofline-limited 400MB part).
// ---------------------------------------------------------------------------
__global__ void add_out(const float* __restrict__ attn,
                        const float* __restrict__ disp,
                        const float* __restrict__ delta,
                        float* __restrict__ out) {
  const size_t i = (size_t)blockIdx.x * blockDim.x + threadIdx.x;  // float4 idx
  const size_t e = i << 2;                   // flat element index into out
  const int n = (int)(e & (NN - 1));
  const size_t rest = e >> 15;               // b*8*48 + a*48 + dd
  const size_t ba = rest / 48;               // b*8 + a
  const int b = (int)(ba >> 3);

  const float4 av = *reinterpret_cast<const float4*>(attn + e);
  const float4 dg = *reinterpret_cast<const float4*>(disp + (size_t)b * NN + n);
  const float4 dl = *reinterpret_cast<const float4*>(delta + ba * NN + n);

  float4 r;
  r.x = av.x + (dg.x >= 2.0f ? 0.0f : dl.x);
  r.y = av.y + (dg.y >= 2.0f ? 0.0f : dl.y);
  r.z = av.z + (dg.z >= 2.0f ? 0.0f : dl.z);
  r.w = av.w + (dg.w >= 2.0f ? 0.0f : dl.w);
  *reinterpret_cast<float4*>(out + e) = r;
}

// ---------------------------------------------------------------------------
extern "C" void kernel_launch(void* const* d_in, const int* in_sizes, int n_in,
                              void* d_out, int out_size, void* d_ws, size_t ws_size,
                              hipStream_t stream) {
  const float* attn = (const float*)d_in[0];  // (4, 8, 48, 128, 256)
  const float* feat = (const float*)d_in[1];  // (4, 80, 128, 256)
  const float* disp = (const float*)d_in[2];  // (4, 1, 128, 256)
  const float* Wq   = (const float*)d_in[3];  // (64, 80)
  const float* Wk   = (const float*)d_in[4];
  const float* Wv   = (const float*)d_in[5];
  const float* Wo   = (const float*)d_in[6];  // (8, 64)
  float* out = (float*)d_out;

  // workspace carve-up: Q, K, V (each B*EMB*NN f32), delta (B*8*NN f32)
  float* Q     = (float*)d_ws;
  float* K     = Q + (size_t)BB * EMB * NN;
  float* V     = K + (size_t)BB * EMB * NN;
  float* delta = V + (size_t)BB * EMB * NN;

  // Stage 1: 4 batches * 3 matrices * 4 e-tiles * 2048 n-tiles = 98304 waves
  //          = 12288 blocks of 256 threads (8 wave32 each)
  qkv_wmma<<<12288, 256, 0, stream>>>(feat, Wq, Wk, Wv, Q, K, V);

  // Stage 2: one thread per pixel: 4*32768 = 131072 threads
  attn_win<<<512, 256, 0, stream>>>(Q, K, V, Wo, delta);

  // Stage 3: 50331648 / 4 = 12582912 float4s -> 49152 blocks of 256
  add_out<<<49152, 256, 0, stream>>>(attn, disp, delta, out);
}